// FixedPastCausalAttention_42872363548849
// MI455X (gfx1250) — compile-verified
//
#include <hip/hip_runtime.h>
#include <math.h>

#define NB 2
#define LL 2048
#define DD 1024
#define HH 16
#define DH 64
#define NHD (NB*HH)            // 32 (n,h) pairs
#define INV_SCALE (1.0f/32.0f) // 1/sqrt(D)
#define NEG_BIG (-3.0e18f)     // (-1e20)/32, exp underflows to exactly 0

typedef __attribute__((ext_vector_type(2))) float v2f;
typedef __attribute__((ext_vector_type(8))) float v8f;

__device__ __forceinline__ v8f wmma_f32_4(v2f a, v2f b, v8f c) {
  // D = A(16x4,f32) * B(4x16,f32) + C(16x16,f32); wave32, full precision
  return __builtin_amdgcn_wmma_f32_16x16x4_f32(false, a, false, b, (short)0, c,
                                               false, false);
}

// Load 16 K-chunk fragments (row-major LxDH matrix, rows r0..r0+15).
// Per-lane pattern: X[r0 + (lane&15)][4*kk + 2*(lane>>4) + {0,1}]
__device__ __forceinline__ void load_frags(v2f* f, const float* __restrict__ base,
                                           int r0, int la, int kOff) {
  #pragma unroll
  for (int kk = 0; kk < 16; ++kk) {
    const float* p = base + (size_t)(r0 + la) * DH + kk * 4 + kOff;
    f[kk] = (v2f){p[0], p[1]};
  }
}

// Full 16x16x64 tile: 16 chained f32 WMMAs.
__device__ __forceinline__ v8f tile_mma(const v2f* aF, const v2f* bF) {
  v8f c = {0.f, 0.f, 0.f, 0.f, 0.f, 0.f, 0.f, 0.f};
  #pragma unroll
  for (int kk = 0; kk < 16; ++kk) c = wmma_f32_4(aF[kk], bF[kk], c);
  return c;
}

// ---------------------------------------------------------------------------
// 1) Per-head projection: out[(n*H+h)*L + l][d] = sum_e x[n][l][h*64+e]*W[d][e]
// ---------------------------------------------------------------------------
__global__ void __launch_bounds__(256) proj_kernel(const float* __restrict__ x,
                                                   const float* __restrict__ W,
                                                   float* __restrict__ out) {
  __shared__ float Wl[DH * DH];
  const int tid = threadIdx.x;
  for (int i = tid; i < DH * DH; i += blockDim.x) Wl[i] = W[i];
  __syncthreads();

  const int r   = blockIdx.x * blockDim.x + tid;  // 0 .. N*L*H-1
  const int n   = r / (LL * HH);
  const int rem = r % (LL * HH);
  const int l   = rem / HH;
  const int h   = rem % HH;
  const float* xin = x + ((size_t)(n * LL + l)) * DD + h * DH;
  float* o = out + ((size_t)((n * HH + h) * LL + l)) * DH;

  #pragma unroll
  for (int dc = 0; dc < 4; ++dc) {
    float acc[16];
    #pragma unroll
    for (int j = 0; j < 16; ++j) acc[j] = 0.f;
    #pragma unroll 8
    for (int e = 0; e < DH; ++e) {
      const float xe = xin[e];
      #pragma unroll
      for (int j = 0; j < 16; ++j) acc[j] += xe * Wl[(dc * 16 + j) * DH + e];
    }
    #pragma unroll
    for (int j = 0; j < 16; ++j) o[dc * 16 + j] = acc[j];
  }
}

// ---------------------------------------------------------------------------
// 2) samp_energy: se[nh*L + l] = dot(Qp[nh][l], Kp[nh][l])
// ---------------------------------------------------------------------------
__global__ void __launch_bounds__(256) se_kernel(const float* __restrict__ Qp,
                                                 const float* __restrict__ Kp,
                                                 float* __restrict__ se) {
  const int i = blockIdx.x * blockDim.x + threadIdx.x;  // < NHD*LL
  const float* q = Qp + (size_t)i * DH;
  const float* k = Kp + (size_t)i * DH;
  float s = 0.f;
  #pragma unroll 16
  for (int d = 0; d < DH; ++d) s += q[d] * k[d];
  se[i] = s;
}

// ---------------------------------------------------------------------------
// 3) Flash pass 1: per-row (max, sumexp). One wave per 32-row DOUBLE strip
//    (strips 2p, 2p+1): every streamed B tile feeds two tile-MMAs (2x
//    arithmetic intensity vs one strip). 2p mask-free main tiles (even ->
//    clean ping-pong), two peeled diag epilogue tiles (branch-free selects).
// ---------------------------------------------------------------------------
__global__ void __launch_bounds__(256) rowstats_kernel(
    const float* __restrict__ Qp, const float* __restrict__ OKp,
    const float* __restrict__ se, float* __restrict__ rowm,
    float* __restrict__ rowz, float* __restrict__ colsum) {
  const int lane = threadIdx.x & 31;
  const int wid  = blockIdx.x * (blockDim.x >> 5) + (threadIdx.x >> 5);
  const int nh = wid >> 6;       // 64 strip-pairs per (n,h)
  const int p  = wid & 63;
  const int q0 = p << 5;         // rows q0 .. q0+31
  const int P  = p << 1;         // # of both-strip (mask-free) tiles
  const int la = lane & 15;
  const int hi = lane >> 4;
  const int kOff = hi * 2;
  const float* Q  = Qp  + (size_t)nh * LL * DH;
  const float* OK = OKp + (size_t)nh * LL * DH;

  v2f aF0[16], aF1[16];
  load_frags(aF0, Q, q0, la, kOff);
  load_frags(aF1, Q, q0 + 16, la, kOff);

  float seA[8], seB[8];
  #pragma unroll
  for (int r2 = 0; r2 < 8; ++r2) {
    seA[r2] = se[nh * LL + q0 + r2 + 8 * hi] * INV_SCALE;
    seB[r2] = se[nh * LL + q0 + 16 + r2 + 8 * hi] * INV_SCALE;
  }

  float mA[8], sA[8], mB[8], sB[8];
  #pragma unroll
  for (int r2 = 0; r2 < 8; ++r2) {
    mA[r2] = -INFINITY; sA[r2] = 0.f;
    mB[r2] = -INFINITY; sB[r2] = 0.f;
  }

  auto upd = [&](float* m, float* s, const v8f& c) {
    #pragma unroll
    for (int r2 = 0; r2 < 8; ++r2) {
      const float e = c[r2] * INV_SCALE;
      const float nm = fmaxf(m[r2], e);
      s[r2] = s[r2] * __expf(m[r2] - nm) + __expf(e - nm);
      m[r2] = nm;
    }
  };
  auto updDiag = [&](float* m, float* s, const v8f& c, const float* seR) {
    #pragma unroll
    for (int r2 = 0; r2 < 8; ++r2) {
      const int rr = r2 + 8 * hi;            // row within tile; col = la
      float e = c[r2] * INV_SCALE;
      e = (la > rr) ? NEG_BIG : e;           // future -> masked
      e = (la == rr) ? seR[r2] : e;          // diagonal swap
      const float nm = fmaxf(m[r2], e);
      s[r2] = s[r2] * __expf(m[r2] - nm) + __expf(e - nm);
      m[r2] = nm;
    }
  };

  v2f b0[16], b1[16];
  load_frags(b0, OK, 0, la, kOff);
  for (int t = 0; t + 1 < P; t += 2) {       // P is even: no leftover
    load_frags(b1, OK, (t + 1) << 4, la, kOff);
    __builtin_prefetch(OK + (size_t)(((t + 3) << 4) + la) * DH + kOff, 0, 3);
    upd(mA, sA, tile_mma(aF0, b0));
    upd(mB, sB, tile_mma(aF1, b0));
    load_frags(b0, OK, (t + 2) << 4, la, kOff);
    upd(mA, sA, tile_mma(aF0, b1));
    upd(mB, sB, tile_mma(aF1, b1));
  }
  // epilogue: tile P (k0=q0): diag for strip0, fully valid for strip1;
  //           tile P+1 (k0=q0+16): diag for strip1 (strip0 fully masked).
  load_frags(b1, OK, (P + 1) << 4, la, kOff);
  updDiag(mA, sA, tile_mma(aF0, b0), seA);
  upd(mB, sB, tile_mma(aF1, b0));
  updDiag(mB, sB, tile_mma(aF1, b1), seB);

  // reduce (m,s) across the 16 lanes of each half-wave
  #pragma unroll
  for (int r2 = 0; r2 < 8; ++r2) {
    #pragma unroll
    for (int off = 1; off < 16; off <<= 1) {
      float om = __shfl_xor(mA[r2], off, 32);
      float os = __shfl_xor(sA[r2], off, 32);
      float nm = fmaxf(mA[r2], om);
      sA[r2] = sA[r2] * __expf(mA[r2] - nm) + os * __expf(om - nm);
      mA[r2] = nm;
      om = __shfl_xor(mB[r2], off, 32);
      os = __shfl_xor(sB[r2], off, 32);
      nm = fmaxf(mB[r2], om);
      sB[r2] = sB[r2] * __expf(mB[r2] - nm) + os * __expf(om - nm);
      mB[r2] = nm;
    }
  }
  if (la == 0) {
    #pragma unroll
    for (int r2 = 0; r2 < 8; ++r2) {
      const int rowA = q0 + r2 + 8 * hi;
      const int rowB = rowA + 16;
      rowm[nh * LL + rowA] = mA[r2];
      rowz[nh * LL + rowA] = sA[r2];
      colsum[nh * LL + rowA] = 0.f;   // zero-init for pass 2 atomics
      rowm[nh * LL + rowB] = mB[r2];
      rowz[nh * LL + rowB] = sB[r2];
      colsum[nh * LL + rowB] = 0.f;
    }
  }
}

// ---------------------------------------------------------------------------
// 4) Flash pass 2: recompute tiles, p = exp(e - m)/Z, accumulate column sums
//    (diag excluded). Same double-strip + peel + ping-pong structure; one
//    atomic per column per tile (both strips' partials pre-summed).
// ---------------------------------------------------------------------------
__global__ void __launch_bounds__(256) colsum_kernel(
    const float* __restrict__ Qp, const float* __restrict__ OKp,
    const float* __restrict__ rowm, const float* __restrict__ rowz,
    float* __restrict__ colsum) {
  const int lane = threadIdx.x & 31;
  const int wid  = blockIdx.x * (blockDim.x >> 5) + (threadIdx.x >> 5);
  const int nh = wid >> 6;
  const int p  = wid & 63;
  const int q0 = p << 5;
  const int P  = p << 1;
  const int la = lane & 15;
  const int hi = lane >> 4;
  const int kOff = hi * 2;
  const float* Q  = Qp  + (size_t)nh * LL * DH;
  const float* OK = OKp + (size_t)nh * LL * DH;

  v2f aF0[16], aF1[16];
  load_frags(aF0, Q, q0, la, kOff);
  load_frags(aF1, Q, q0 + 16, la, kOff);

  float rmA[8], rzA[8], rmB[8], rzB[8];
  #pragma unroll
  for (int r2 = 0; r2 < 8; ++r2) {
    const int rowA = q0 + r2 + 8 * hi;
    rmA[r2] = rowm[nh * LL + rowA];
    rzA[r2] = 1.0f / rowz[nh * LL + rowA];
    rmB[r2] = rowm[nh * LL + rowA + 16];
    rzB[r2] = 1.0f / rowz[nh * LL + rowA + 16];
  }

  auto tsum = [&](const v8f& c, const float* rm, const float* rz) -> float {
    float ls = 0.f;
    #pragma unroll
    for (int r2 = 0; r2 < 8; ++r2)
      ls += __expf(c[r2] * INV_SCALE - rm[r2]) * rz[r2];
    return ls;
  };
  auto tsumDiag = [&](const v8f& c, const float* rm, const float* rz) -> float {
    float ls = 0.f;
    #pragma unroll
    for (int r2 = 0; r2 < 8; ++r2) {
      const int rr = r2 + 8 * hi;
      float pv = __expf(c[r2] * INV_SCALE - rm[r2]) * rz[r2];
      ls += (la >= rr) ? 0.f : pv;   // masked future + excluded diagonal
    }
    return ls;
  };
  auto emit = [&](float ls, int k0) {
    ls += __shfl_xor(ls, 16, 32);    // combine row-halves
    if (hi == 0) atomicAdd(&colsum[nh * LL + k0 + la], ls);
  };

  v2f b0[16], b1[16];
  load_frags(b0, OK, 0, la, kOff);
  for (int t = 0; t + 1 < P; t += 2) {
    load_frags(b1, OK, (t + 1) << 4, la, kOff);
    __builtin_prefetch(OK + (size_t)(((t + 3) << 4) + la) * DH + kOff, 0, 3);
    emit(tsum(tile_mma(aF0, b0), rmA, rzA) +
         tsum(tile_mma(aF1, b0), rmB, rzB), t << 4);
    load_frags(b0, OK, (t + 2) << 4, la, kOff);
    emit(tsum(tile_mma(aF0, b1), rmA, rzA) +
         tsum(tile_mma(aF1, b1), rmB, rzB), (t + 1) << 4);
  }
  // epilogue tiles
  load_frags(b1, OK, (P + 1) << 4, la, kOff);
  emit(tsumDiag(tile_mma(aF0, b0), rmA, rzA) +
       tsum(tile_mma(aF1, b0), rmB, rzB), q0);
  emit(tsumDiag(tile_mma(aF1, b1), rmB, rzB), q0 + 16);
}

// ---------------------------------------------------------------------------
// 5) Combine: attn[n][l][h*64+d] = diag_p * v + colsum * ov
// ---------------------------------------------------------------------------
__global__ void __launch_bounds__(256) combine_kernel(
    const float* __restrict__ Vp, const float* __restrict__ OVp,
    const float* __restrict__ se, const float* __restrict__ rowm,
    const float* __restrict__ rowz, const float* __restrict__ colsum,
    float* __restrict__ attn) {
  const int i = blockIdx.x * blockDim.x + threadIdx.x;  // < NHD*LL
  const int nh = i / LL, l = i % LL;
  const int n = nh / HH, h = nh % HH;
  const float dp = __expf(se[i] * INV_SCALE - rowm[i]) / rowz[i];
  const float cs = colsum[i];
  const float* v  = Vp  + (size_t)i * DH;
  const float* ov = OVp + (size_t)i * DH;
  float* o = attn + ((size_t)(n * LL + l)) * DD + h * DH;
  #pragma unroll 8
  for (int d = 0; d < DH; ++d) o[d] = dp * v[d] + cs * ov[d];
}

// ---------------------------------------------------------------------------
// 6) Final FC: out(4096x1024) = attn @ fc_w^T + fc_b.
//    Fully unrolled 16-chunk (64-K each) ping-pong; staged frag loads.
// ---------------------------------------------------------------------------
__global__ void __launch_bounds__(256) fc_kernel(
    const float* __restrict__ A, const float* __restrict__ Wfc,
    const float* __restrict__ bias, float* __restrict__ out) {
  const int lane = threadIdx.x & 31;
  const int wid  = blockIdx.x * (blockDim.x >> 5) + (threadIdx.x >> 5);
  const int TN = DD / 16;                  // 64 N-tiles
  const int m0 = (wid / TN) * 16;
  const int j0 = (wid % TN) * 16;
  const int la = lane & 15, hi = lane >> 4, kOff = hi * 2;

  auto loadAB = [&](v2f* a, v2f* b, int ch) {
    #pragma unroll
    for (int kk = 0; kk < 16; ++kk) {
      const float* pa = A   + (size_t)(m0 + la) * DD + ch * 64 + kk * 4 + kOff;
      const float* pb = Wfc + (size_t)(j0 + la) * DD + ch * 64 + kk * 4 + kOff;
      a[kk] = (v2f){pa[0], pa[1]};
      b[kk] = (v2f){pb[0], pb[1]};
    }
  };
  auto accum = [&](v8f c, const v2f* a, const v2f* b) {
    #pragma unroll
    for (int kk = 0; kk < 16; ++kk) c = wmma_f32_4(a[kk], b[kk], c);
    return c;
  };

  v2f a0[16], b0[16], a1[16], b1[16];
  v8f c = {0.f, 0.f, 0.f, 0.f, 0.f, 0.f, 0.f, 0.f};
  loadAB(a0, b0, 0);
  #pragma unroll
  for (int t = 0; t < 16; t += 2) {        // 16 chunks of K=64
    loadAB(a1, b1, t + 1);
    c = accum(c, a0, b0);
    if (t + 2 < 16) loadAB(a0, b0, t + 2);
    c = accum(c, a1, b1);
  }

  const int col = j0 + la;
  const float bv = bias[col];
  #pragma unroll
  for (int r2 = 0; r2 < 8; ++r2) {
    const int row = m0 + r2 + 8 * hi;
    out[(size_t)row * DD + col] = c[r2] + bv;
  }
}

// ---------------------------------------------------------------------------
extern "C" void kernel_launch(void* const* d_in, const int* in_sizes, int n_in,
                              void* d_out, int out_size, void* d_ws,
                              size_t ws_size, hipStream_t stream) {
  (void)in_sizes; (void)n_in; (void)out_size; (void)ws_size;
  const float* values        = (const float*)d_in[0];
  const float* keys          = (const float*)d_in[1];
  const float* query         = (const float*)d_in[2];
  const float* origin_values = (const float*)d_in[3];
  const float* origin_keys   = (const float*)d_in[4];
  const float* Wv            = (const float*)d_in[5];
  const float* Wk            = (const float*)d_in[6];
  const float* Wq            = (const float*)d_in[7];
  const float* fc_w          = (const float*)d_in[8];
  const float* fc_b          = (const float*)d_in[9];
  // d_in[10] = mask: pure causal for these shapes, handled analytically.

  float* ws = (float*)d_ws;
  const size_t MSZ = (size_t)NHD * LL * DH;  // 4,194,304 floats per matrix
  float* Qp  = ws + 0 * MSZ;
  float* Kp  = ws + 1 * MSZ;
  float* Vp  = ws + 2 * MSZ;
  float* OKp = ws + 3 * MSZ;
  float* OVp = ws + 4 * MSZ;
  float* se     = ws + 5 * MSZ;
  float* rowm   = se   + (size_t)NHD * LL;
  float* rowz   = rowm + (size_t)NHD * LL;
  float* colsum = rowz + (size_t)NHD * LL;
  float* attn   = colsum + (size_t)NHD * LL;  // N*L*D floats
  float* out = (float*)d_out;

  const dim3 blk(256);
  proj_kernel<<<256, blk, 0, stream>>>(query,         Wq, Qp);
  proj_kernel<<<256, blk, 0, stream>>>(keys,          Wk, Kp);
  proj_kernel<<<256, blk, 0, stream>>>(values,        Wv, Vp);
  proj_kernel<<<256, blk, 0, stream>>>(origin_keys,   Wk, OKp);
  proj_kernel<<<256, blk, 0, stream>>>(origin_values, Wv, OVp);
  se_kernel<<<256, blk, 0, stream>>>(Qp, Kp, se);
  rowstats_kernel<<<256, blk, 0, stream>>>(Qp, OKp, se, rowm, rowz, colsum);
  colsum_kernel<<<256, blk, 0, stream>>>(Qp, OKp, rowm, rowz, colsum);
  combine_kernel<<<256, blk, 0, stream>>>(Vp, OVp, se, rowm, rowz, colsum, attn);
  fc_kernel<<<2048, blk, 0, stream>>>(attn, fc_w, fc_b, out);
}